// SpGAT_37039797960752
// MI455X (gfx1250) — compile-verified
//
#include <hip/hip_runtime.h>
#include <hip/hip_bf16.h>
#include <math.h>

// Problem constants (from reference)
#define GAT_N      50000
#define GAT_E      1600000
#define GAT_H      8
#define GAT_ALPHA  0.2f

typedef float v2f __attribute__((ext_vector_type(2)));
typedef float v8f __attribute__((ext_vector_type(8)));

#if defined(__AMDGCN__) && __has_builtin(__builtin_amdgcn_wmma_f32_16x16x4_f32)
#define GAT_HAVE_WMMA 1
#endif

// ---------------------------------------------------------------------------
// WMMA fp32 GEMM: C[Nr,Mo] = act(A[Nr,K] @ B[K,Mo] + bias)
// One wave32 computes one 16x16 tile of C via V_WMMA_F32_16X16X4_F32.
// Nr multiple of 16 (50000 = 3125*16); K multiple of 4; Mo guarded at stores.
// Column guard: WMMA column n only depends on B column n (lanes n / n+16),
// so out-of-range lanes just CLAMP their B address (duplicate col Mo-1); the
// resulting garbage C columns are never stored. Inner loop is branchless.
// Fragment layout per ISA 7.12.2:
//   A 16x4 (2 VGPRs): lane r (<16) -> row r, K={0,1}; lane r+16 -> row r, K={2,3}
//   B 4x16 (2 VGPRs): mirrored (lane r -> col r)
//   C 16x16 (8 VGPRs): acc[v] = C[half*8+v][lane&15]
// ---------------------------------------------------------------------------
__global__ void k_gemm(const float* __restrict__ A, const float* __restrict__ B,
                       const float* __restrict__ bias, float* __restrict__ C,
                       int K, int Mo, int ldA, int ldB, int ldC, int act,
                       int tiles, int tiles_m)
{
    int wave = blockIdx.x * (blockDim.x >> 5) + (threadIdx.x >> 5);
    if (wave >= tiles) return;                 // wave-uniform: EXEC stays all-1s
    int tr   = wave / tiles_m;
    int tc   = wave - tr * tiles_m;
    int lane = threadIdx.x & 31;
    int r    = lane & 15;
    int half = lane >> 4;
    int arow = tr * 16 + r;
    int col  = tc * 16 + r;
    bool colOK = (col < Mo);
    int colc  = colOK ? col : (Mo - 1);        // clamped: in-bounds, data unused

    const float* Ap = A + (long long)arow * ldA + 2 * half;
    const float* Bp = B + (long long)(2 * half) * ldB + colc;
    const long long bstep = (long long)4 * ldB;

    v8f acc = {0.f, 0.f, 0.f, 0.f, 0.f, 0.f, 0.f, 0.f};

#ifdef GAT_HAVE_WMMA
    #pragma unroll 2
    for (int k0 = 0; k0 < K; k0 += 4) {
        __builtin_prefetch(Ap + 16, 0, 3);     // near-temporal prefetch on A stream
        v2f af; af.x = Ap[0];   af.y = Ap[1];
        v2f bf; bf.x = Bp[0];   bf.y = Bp[ldB];
        acc = __builtin_amdgcn_wmma_f32_16x16x4_f32(
                  false, af, false, bf, (short)0, acc, false, false);
        Ap += 4;
        Bp += bstep;
    }
#else
    {   // scalar fallback (host pass / builtin missing) — same math
        int m0 = tr * 16 + half * 8;
        for (int k = 0; k < K; ++k) {
            float bb = colOK ? B[(long long)k * ldB + col] : 0.f;
            #pragma unroll
            for (int v = 0; v < 8; ++v)
                acc[v] += A[(long long)(m0 + v) * ldA + k] * bb;
        }
    }
#endif

    if (colOK) {
        float bv = bias ? bias[col] : 0.f;
        int m0 = tr * 16 + half * 8;
        #pragma unroll
        for (int v = 0; v < 8; ++v) {
            float val = acc[v] + bv;
            if (act == 1) val = val > 0.f ? val : 0.f;   // relu
            C[(long long)(m0 + v) * ldC + col] = val;
        }
    }
}

// ---------------------------------------------------------------------------
// Attention projections: AS[n,h] = Wh[n,h*64:]·a_h[0:64], AD with a_h[64:128]
// ---------------------------------------------------------------------------
__global__ void k_attn_proj(const float* __restrict__ Wh, int ldWh,
                            const float* __restrict__ a, int aStride, int H_,
                            float* __restrict__ AS, float* __restrict__ AD, int Nn)
{
    int t = blockIdx.x * blockDim.x + threadIdx.x;
    if (t >= Nn * H_) return;
    int n = t / H_, hd = t - n * H_;
    const float* w  = Wh + (long long)n * ldWh + hd * 64;
    const float* ah = a + hd * aStride;
    float s0 = 0.f, s1 = 0.f;
    #pragma unroll 8
    for (int j = 0; j < 64; ++j) {
        float xv = w[j];
        s0 += xv * ah[j];
        s1 += xv * ah[64 + j];
    }
    AS[t] = s0;
    AD[t] = s1;
}

// ordered-uint encoding so unsigned atomicMax == float max
__device__ __forceinline__ unsigned f32_ord(float f) {
    unsigned u = __float_as_uint(f);
    return (u & 0x80000000u) ? ~u : (u | 0x80000000u);
}
__device__ __forceinline__ float ord_f32(unsigned u) {
    return (u & 0x80000000u) ? __uint_as_float(u & 0x7fffffffu)
                             : __uint_as_float(~u);
}

__global__ void k_fill(unsigned* __restrict__ p, unsigned v, long long n)
{
    long long t = (long long)blockIdx.x * blockDim.x + threadIdx.x;
    if (t < n) p[t] = v;
}

// logits[e,h] = leaky_relu(AS[src]+AD[dst]+ef*a[2d], 0.2); segment-max into M
__global__ void k_edge_logits(const int* __restrict__ src, const int* __restrict__ dst,
                              const float* __restrict__ ef,
                              const float* __restrict__ AS, const float* __restrict__ AD,
                              const float* __restrict__ a, int aStride, int H_,
                              float* __restrict__ ET, unsigned* __restrict__ M_,
                              long long total)
{
    long long t = (long long)blockIdx.x * blockDim.x + threadIdx.x;
    if (t >= total) return;
    int e  = (int)(t / H_);
    int hd = (int)(t - (long long)e * H_);
    int s = src[e], d = dst[e];
    float z = AS[(long long)s * H_ + hd] + AD[(long long)d * H_ + hd]
            + ef[e] * a[hd * aStride + 128];
    z = z > 0.f ? z : GAT_ALPHA * z;
    ET[t] = z;
    atomicMax(&M_[(long long)d * H_ + hd], f32_ord(z));
}

// ea = exp(logit - m[dst]); segment-sum into S
__global__ void k_edge_exp(const int* __restrict__ dst,
                           float* __restrict__ ET, const unsigned* __restrict__ M_,
                           float* __restrict__ S_, int H_, long long total)
{
    long long t = (long long)blockIdx.x * blockDim.x + threadIdx.x;
    if (t >= total) return;
    int e  = (int)(t / H_);
    int hd = (int)(t - (long long)e * H_);
    int d = dst[e];
    float ea = __expf(ET[t] - ord_f32(M_[(long long)d * H_ + hd]));
    ET[t] = ea;
    atomicAdd(&S_[(long long)d * H_ + hd], ea);
}

// mean over heads of attn[e,h] = ea/(s[dst]+1e-16)
__global__ void k_edge_mean(const int* __restrict__ dst, const float* __restrict__ ET,
                            const float* __restrict__ S_, int H_,
                            float* __restrict__ out, int E_)
{
    int e = blockIdx.x * blockDim.x + threadIdx.x;
    if (e >= E_) return;
    int d = dst[e];
    float s = 0.f;
    for (int hd = 0; hd < H_; ++hd)
        s += ET[(long long)e * H_ + hd] / (S_[(long long)d * H_ + hd] + 1e-16f);
    out[e] = s / (float)H_;
}

// hp[dst, c..c+3] += attn[e, c/64] * Wh[src, c..c+3]  (float4 gather, 4 atomics)
__global__ void k_edge_aggregate4(const int* __restrict__ src, const int* __restrict__ dst,
                                  const float* __restrict__ ET, const float* __restrict__ S_,
                                  const float* __restrict__ Wh, float* __restrict__ HP,
                                  int H_, int D, long long total4)
{
    long long t = (long long)blockIdx.x * blockDim.x + threadIdx.x;
    if (t >= total4) return;
    int q = D >> 2;                     // float4 groups per edge
    int e = (int)(t / q);
    int g = (int)(t - (long long)e * q);
    int c = g << 2;
    int hd = c >> 6;
    int s = src[e], d = dst[e];
    float attn = ET[(long long)e * H_ + hd] / (S_[(long long)d * H_ + hd] + 1e-16f);
    float4 wv = *(const float4*)(Wh + (long long)s * D + c);   // global_load_b128
    float* hp = HP + (long long)d * D + c;
    atomicAdd(hp + 0, attn * wv.x);
    atomicAdd(hp + 1, attn * wv.y);
    atomicAdd(hp + 2, attn * wv.z);
    atomicAdd(hp + 3, attn * wv.w);
}

__global__ void k_elu(float* __restrict__ p, long long n)
{
    long long t = (long long)blockIdx.x * blockDim.x + threadIdx.x;
    if (t >= n) return;
    float x = p[t];
    p[t] = x > 0.f ? x : __expf(x) - 1.f;
}

__global__ void k_softmax10(const float* __restrict__ LG, float* __restrict__ out, int Nn)
{
    int n = blockIdx.x * blockDim.x + threadIdx.x;
    if (n >= Nn) return;
    float v[10];
    float m = -1e30f;
    #pragma unroll
    for (int c = 0; c < 10; ++c) { v[c] = LG[(long long)n * 10 + c]; m = v[c] > m ? v[c] : m; }
    float s = 0.f;
    #pragma unroll
    for (int c = 0; c < 10; ++c) { v[c] = __expf(v[c] - m); s += v[c]; }
    float inv = 1.f / s;
    #pragma unroll
    for (int c = 0; c < 10; ++c) out[(long long)n * 10 + c] = v[c] * inv;
}

// ---------------------------------------------------------------------------
// Host orchestration
// ---------------------------------------------------------------------------
static inline unsigned int gblocks(long long n, int b) {
    return (unsigned int)((n + b - 1) / b);
}

extern "C" void kernel_launch(void* const* d_in, const int* in_sizes, int n_in,
                              void* d_out, int out_size, void* d_ws, size_t ws_size,
                              hipStream_t stream)
{
    (void)in_sizes; (void)n_in; (void)out_size; (void)ws_size;
    const int N = GAT_N, E = GAT_E, H = GAT_H;

    const float* x         = (const float*)d_in[0];
    const int*   edgeA     = (const int*)  d_in[1];
    const int*   edgeB     = (const int*)  d_in[2];
    const float* edge_feat = (const float*)d_in[3];
    const float* Win       = (const float*)d_in[4];
    const float* bin_      = (const float*)d_in[5];
    const float* Wu        = (const float*)d_in[6];
    const float* au        = (const float*)d_in[7];
    const float* Wv        = (const float*)d_in[8];
    const float* av        = (const float*)d_in[9];
    const float* Wo_uv     = (const float*)d_in[10];
    const float* ao_uv     = (const float*)d_in[11];
    const float* Wo_vu     = (const float*)d_in[12];
    const float* ao_vu     = (const float*)d_in[13];
    const float* W1        = (const float*)d_in[14];
    const float* b1        = (const float*)d_in[15];
    const float* W2        = (const float*)d_in[16];
    const float* b2        = (const float*)d_in[17];
    const float* W3        = (const float*)d_in[18];

    const int* srcA = edgeA;        // edgeA[0]
    const int* dstA = edgeA + E;    // edgeA[1]
    const int* srcB = edgeB;
    const int* dstB = edgeB + E;

    float* out_softmax = (float*)d_out;                      // [N,10]
    float* out_ne      = (float*)d_out + (long long)N * 10;  // [E] new_edge_

    // ---- workspace carve-up (~322 MB) ----
    char* wp = (char*)d_ws;
    auto carve = [&](size_t bytes) -> void* {
        void* q = (void*)wp;
        wp += (bytes + 255) & ~(size_t)255;
        return q;
    };
    float*    bufH   = (float*)   carve((size_t)N * 64 * 4);   // h -> later o1
    float*    bufWH  = (float*)   carve((size_t)N * 512 * 4);  // Whu_all / Whv_all
    float*    bufHP  = (float*)   carve((size_t)N * 512 * 4);  // hp/x1 -> hp/x3
    float*    bufET  = (float*)   carve((size_t)E * H * 4);    // logits -> ea
    unsigned* bufM   = (unsigned*)carve((size_t)N * H * 4);    // segment max (ordered)
    float*    bufS   = (float*)   carve((size_t)N * H * 4);    // segment sum
    float*    bufAS  = (float*)   carve((size_t)N * H * 4);    // src-side proj
    float*    bufAD  = (float*)   carve((size_t)N * H * 4);    // dst-side proj
    float*    bufNE  = (float*)   carve((size_t)E * 4);        // new_edge (layer1)
    float*    bufX2  = (float*)   carve((size_t)N * 64 * 4);   // x2 -> later o2
    float*    bufWHS = (float*)   carve((size_t)N * 64 * 4);   // Wh for 1-head GATs
    float*    bufX4  = (float*)   carve((size_t)N * 64 * 4);   // x4
    float*    bufLG  = (float*)   carve((size_t)N * 10 * 4);   // MLP logits

    auto gemm = [&](const float* A, const float* B, const float* bias, float* C,
                    int K, int Mo, int ldA, int ldB, int ldC, int act) {
        int tiles_n = N / 16;                      // N divisible by 16
        int tiles_m = (Mo + 15) / 16;
        int tiles   = tiles_n * tiles_m;
        int blocks  = (tiles + 3) / 4;             // 4 waves / block
        k_gemm<<<blocks, 128, 0, stream>>>(A, B, bias, C, K, Mo, ldA, ldB, ldC,
                                           act, tiles, tiles_m);
    };

    // full GAT layer (edge phase) given precomputed Wh in [N, H_*64]
    auto run_gat = [&](const int* src, const int* dst, const float* ef,
                       const float* Wh, int H_, const float* a, int aStride,
                       float* HPout, float* meanOut, int elu) {
        int D = H_ * 64;
        long long EH = (long long)E * H_;
        long long NH = (long long)N * H_;
        long long ND = (long long)N * D;
        k_attn_proj<<<gblocks(NH, 256), 256, 0, stream>>>(Wh, D, a, aStride, H_,
                                                          bufAS, bufAD, N);
        k_fill<<<gblocks(NH, 256), 256, 0, stream>>>(bufM, 0x007FFFFFu, NH); // ord(-inf)
        k_fill<<<gblocks(NH, 256), 256, 0, stream>>>((unsigned*)bufS, 0u, NH);
        k_fill<<<gblocks(ND, 256), 256, 0, stream>>>((unsigned*)HPout, 0u, ND);
        k_edge_logits<<<gblocks(EH, 256), 256, 0, stream>>>(src, dst, ef, bufAS, bufAD,
                                                            a, aStride, H_, bufET, bufM, EH);
        k_edge_exp<<<gblocks(EH, 256), 256, 0, stream>>>(dst, bufET, bufM, bufS, H_, EH);
        if (meanOut)
            k_edge_mean<<<gblocks(E, 256), 256, 0, stream>>>(dst, bufET, bufS, H_, meanOut, E);
        long long ED4 = (long long)E * (D >> 2);
        k_edge_aggregate4<<<gblocks(ED4, 256), 256, 0, stream>>>(src, dst, bufET, bufS,
                                                                 Wh, HPout, H_, D, ED4);
        if (elu)
            k_elu<<<gblocks(ND, 256), 256, 0, stream>>>(HPout, ND);
    };

    // ---- stage 0: h = x @ Win + bin_ ----
    gemm(x, Win, bin_, bufH, 128, 64, 128, 64, 64, 0);

    // ---- stage 1: 8-head GAT on edgeA (h -> x1) ----
    for (int hd = 0; hd < H; ++hd)
        gemm(bufH, Wu + (size_t)hd * 64 * 64, nullptr, bufWH + hd * 64,
             64, 64, 64, 64, 512, 0);
    run_gat(srcA, dstA, edge_feat, bufWH, H, au, 129, bufHP, bufNE, 1);

    // ---- stage 2: 1-head GAT on edgeA (x1 -> x2, no activation) ----
    gemm(bufHP, Wo_uv, nullptr, bufWHS, 512, 64, 512, 64, 64, 0);
    run_gat(srcA, dstA, edge_feat, bufWHS, 1, ao_uv, 129, bufX2, nullptr, 0);

    // ---- stage 3: 8-head GAT on edgeB, ef = new_edge (x2 -> x3) ----
    for (int hd = 0; hd < H; ++hd)
        gemm(bufX2, Wv + (size_t)hd * 64 * 64, nullptr, bufWH + hd * 64,
             64, 64, 64, 64, 512, 0);
    run_gat(srcB, dstB, bufNE, bufWH, H, av, 129, bufHP, out_ne, 1); // mean -> output

    // ---- stage 4: 1-head GAT on edgeB, ef = new_edge (x3 -> x4) ----
    gemm(bufHP, Wo_vu, nullptr, bufWHS, 512, 64, 512, 64, 64, 0);
    run_gat(srcB, dstB, bufNE, bufWHS, 1, ao_vu, 129, bufX4, nullptr, 0);

    // ---- stage 5: MLP + softmax ----
    gemm(bufX4, W1, b1, bufH, 64, 64, 64, 64, 64, 1);   // o1 = relu(x4@W1+b1)
    gemm(bufH, W2, b2, bufX2, 64, 64, 64, 64, 64, 1);   // o2 = relu(o1@W2+b2)
    gemm(bufX2, W3, nullptr, bufLG, 64, 10, 64, 10, 10, 0);
    k_softmax10<<<gblocks(N, 256), 256, 0, stream>>>(bufLG, out_softmax, N);
}